// Model_39676907882997
// MI455X (gfx1250) — compile-verified
//
#include <hip/hip_runtime.h>
#include <hip/hip_bf16.h>

// Single-wave fp32 WMMA kernel:
//   out = in2@t1 + in3@in4 + in5@in6 + in7@in4 + in6@in3,  t1 = in1 with [1,1]=PI
// 7x7 fp32 matrices padded into one 16x16 WMMA tile, K padded to 8
// (2 chunks of K=4) -> 10 v_wmma_f32_16x16x4_f32 in a single wave32.
//
// Padding strategy (minimal masking):
//  - A rows m>=7 / B cols n>=7 only affect output rows/cols that are never
//    stored -> no value masking, just clamped in-bounds addresses.
//  - Only K=7 must vanish from the dot product; zeroing the B side alone is
//    sufficient (A side may hold finite garbage). That is 4 cndmasks total
//    (one per B operand, kc=1 / j=1 / half=1) + 1 for the PI patch.

typedef __attribute__((ext_vector_type(2))) float v2f;
typedef __attribute__((ext_vector_type(8))) float v8f;

#define PI_F 3.1415926f

__global__ void __launch_bounds__(32)
sum5_matmul_wmma_f32(const float* __restrict__ in1, const float* __restrict__ in2,
                     const float* __restrict__ in3, const float* __restrict__ in4,
                     const float* __restrict__ in5, const float* __restrict__ in6,
                     const float* __restrict__ in7, float* __restrict__ out) {
    const int lane = threadIdx.x;          // one wave32; EXEC all-1s at every WMMA
    const int half = (lane >> 4) & 1;      // & 1 gives the compiler the [0,1] range
    const int mn   = lane & 15;            // row for A, col for B/C/D
    const int rc   = mn < 7 ? mn : 6;      // clamped row/col for in-bounds addresses
    const int aro  = rc * 7;               // A row base (element index)

    v8f c = {};

#pragma unroll
    for (int kc = 0; kc < 2; ++kc) {
        // K indices for this chunk (depend only on half -> shared by all matrices).
        const int k0   = 4 * kc + 2 * half;         // j=0: {0,2} or {4,6}, always < 7
        const int k1v  = k0 + 1;                    // j=1: may be 7 (kc=1, half=1)
        const int k1   = k1v < 7 ? k1v : 6;         // clamped address for the k=7 slot
        const bool k1ok = (k1v < 7);                // compile-time true for kc=0

        // A operands (16x4): lane m holds A[m][k0], A[m][k1]; k=7 slot is
        // harmless garbage because the matching B row is zeroed.
        v2f a2, a3, a5, a6, a7;
        a2[0] = in2[aro + k0]; a2[1] = in2[aro + k1];
        a3[0] = in3[aro + k0]; a3[1] = in3[aro + k1];
        a5[0] = in5[aro + k0]; a5[1] = in5[aro + k1];
        a6[0] = in6[aro + k0]; a6[1] = in6[aro + k1];
        a7[0] = in7[aro + k0]; a7[1] = in7[aro + k1];

        // B operands (4x16): unconditional clamped loads, then select-to-zero
        // only the k=7 element (no exec-mask branches).
        const int b0i = k0 * 7 + rc;
        const int b1i = k1 * 7 + rc;
        float v1a = in1[b0i], v1b = in1[b1i];
        float v3a = in3[b0i], v3b = in3[b1i];
        float v4a = in4[b0i], v4b = in4[b1i];
        float v6a = in6[b0i], v6b = in6[b1i];

        v2f b1, b3, b4, b6;
        b1[0] = v1a; b1[1] = k1ok ? v1b : 0.0f;
        b3[0] = v3a; b3[1] = k1ok ? v3b : 0.0f;
        b4[0] = v4a; b4[1] = k1ok ? v4b : 0.0f;
        b6[0] = v6a; b6[1] = k1ok ? v6b : 0.0f;

        // t1 = in1 with [1,1] = PI: element (k=1, n=1) lives at kc=0, j=1,
        // half=0, lane n=1.
        if (kc == 0) {
            const bool patch = (half == 0) && (mn == 1);
            b1[1] = patch ? PI_F : b1[1];
        }

        c = __builtin_amdgcn_wmma_f32_16x16x4_f32(false, a2, false, b1, (short)0, c, false, false); // t2
        c = __builtin_amdgcn_wmma_f32_16x16x4_f32(false, a3, false, b4, (short)0, c, false, false); // t7
        c = __builtin_amdgcn_wmma_f32_16x16x4_f32(false, a5, false, b6, (short)0, c, false, false); // t8
        c = __builtin_amdgcn_wmma_f32_16x16x4_f32(false, a7, false, b4, (short)0, c, false, false); // t12
        c = __builtin_amdgcn_wmma_f32_16x16x4_f32(false, a6, false, b3, (short)0, c, false, false); // t13
    }

    // C/D 16x16 f32 layout: VGPR i -> row M = i + 8*half, col N = lane&15.
    // Only half==0 lanes with n<7 carry stored rows (VGPRs 0..6).
    if (half == 0 && mn < 7) {
#pragma unroll
        for (int i = 0; i < 7; ++i) {
            out[i * 7 + mn] = c[i];
        }
    }
}

extern "C" void kernel_launch(void* const* d_in, const int* in_sizes, int n_in,
                              void* d_out, int out_size, void* d_ws, size_t ws_size,
                              hipStream_t stream) {
    (void)in_sizes; (void)n_in; (void)out_size; (void)d_ws; (void)ws_size;
    const float* in1 = (const float*)d_in[0];
    const float* in2 = (const float*)d_in[1];
    const float* in3 = (const float*)d_in[2];
    const float* in4 = (const float*)d_in[3];
    const float* in5 = (const float*)d_in[4];
    const float* in6 = (const float*)d_in[5];
    const float* in7 = (const float*)d_in[6];
    float* out = (float*)d_out;

    // One wave32 does the entire problem: 10 fp32 WMMAs, ~1.6 KB of traffic.
    sum5_matmul_wmma_f32<<<1, 32, 0, stream>>>(in1, in2, in3, in4, in5, in6, in7, out);
}